// ModelNew_60713657697009
// MI455X (gfx1250) — compile-verified
//
#include <hip/hip_runtime.h>
#include <hip/hip_bf16.h>
#include <stdint.h>

// Problem constants (from reference file).
#define NNZ      4194304     // 2^22 nonzeros
#define DENSE    16          // floats per row (64 bytes)
#define DIMV     8192        // dim0 == dim1 == 8192
#define KEYSPACE 67108864    // 2^26 flat keys
#define WORDS    2097152     // KEYSPACE / 32 bitmap words
#define SCAN_CHUNK 1024      // words per scan block
#define SCAN_BLKS  2048      // WORDS / SCAN_CHUNK

// ---------------------------------------------------------------------------
// Block-wide exclusive scan over 256 unsigned values (Hillis-Steele in LDS).
// ---------------------------------------------------------------------------
__device__ __forceinline__ unsigned block_excl_scan_256(unsigned v, unsigned* lds) {
    const int tid = threadIdx.x;
    lds[tid] = v;
    __syncthreads();
#pragma unroll
    for (int off = 1; off < 256; off <<= 1) {
        unsigned x = (tid >= off) ? lds[tid - off] : 0u;
        __syncthreads();
        lds[tid] += x;
        __syncthreads();
    }
    return lds[tid] - v;
}

// ---------------------------------------------------------------------------
// K1: clear presence + duplicate bitmaps (2*WORDS u32, contiguous).
// ---------------------------------------------------------------------------
__global__ __launch_bounds__(256) void k_clear(unsigned* bm) {
    size_t i = ((size_t)blockIdx.x * 256 + threadIdx.x) * 4;
#pragma unroll
    for (int j = 0; j < 4; ++j) bm[i + j] = 0u;
}

// ---------------------------------------------------------------------------
// K2: mark presence bitmap; atomicOr's old value tells us the key was already
// present -> set the duplicate bit (set iff multiplicity >= 2; deterministic).
// ---------------------------------------------------------------------------
__global__ __launch_bounds__(256) void k_mark(const int* __restrict__ idx0,
                                              const int* __restrict__ idx1,
                                              unsigned* __restrict__ present,
                                              unsigned* __restrict__ dup) {
    size_t i = (size_t)blockIdx.x * 256 + threadIdx.x;
    unsigned k   = (unsigned)idx0[i] * (unsigned)DIMV + (unsigned)idx1[i];
    unsigned w   = k >> 5;
    unsigned bit = 1u << (k & 31u);
    unsigned old = atomicOr(&present[w], bit);
    if (old & bit) atomicOr(&dup[w], bit);
}

// ---------------------------------------------------------------------------
// K3a: per-block popcount partial sums (SCAN_CHUNK words per block).
// ---------------------------------------------------------------------------
__global__ __launch_bounds__(256) void k_scan_partial(const unsigned* __restrict__ present,
                                                      unsigned* __restrict__ blockSums) {
    __shared__ unsigned lds[256];
    size_t base = (size_t)blockIdx.x * SCAN_CHUNK + (size_t)threadIdx.x * 4;
    unsigned s = 0;
#pragma unroll
    for (int j = 0; j < 4; ++j) s += __popc(present[base + j]);
    unsigned excl = block_excl_scan_256(s, lds);
    if (threadIdx.x == 255) blockSums[blockIdx.x] = excl + s;  // block total
}

// ---------------------------------------------------------------------------
// K3b: single-block exclusive scan over SCAN_BLKS partial sums; emit per-block
// bases, n_unique to workspace, and the scalar n_unique output.
// ---------------------------------------------------------------------------
__global__ __launch_bounds__(256) void k_scan_top(const unsigned* __restrict__ blockSums,
                                                  unsigned* __restrict__ blockBase,
                                                  unsigned* __restrict__ nUnique,
                                                  int* __restrict__ outN) {
    __shared__ unsigned lds[256];
    const int tid = threadIdx.x;
    unsigned v[SCAN_BLKS / 256];
    unsigned s = 0;
#pragma unroll
    for (int j = 0; j < SCAN_BLKS / 256; ++j) {
        v[j] = blockSums[tid * (SCAN_BLKS / 256) + j];
        s += v[j];
    }
    unsigned run = block_excl_scan_256(s, lds);
#pragma unroll
    for (int j = 0; j < SCAN_BLKS / 256; ++j) {
        blockBase[tid * (SCAN_BLKS / 256) + j] = run;
        run += v[j];
    }
    if (tid == 255) { *nUnique = run; *outN = (int)run; }
}

// ---------------------------------------------------------------------------
// K3c: word-level exclusive prefix popcount (rank base per bitmap word).
// ---------------------------------------------------------------------------
__global__ __launch_bounds__(256) void k_scan_down(const unsigned* __restrict__ present,
                                                   const unsigned* __restrict__ blockBase,
                                                   unsigned* __restrict__ wordPrefix) {
    __shared__ unsigned lds[256];
    size_t base = (size_t)blockIdx.x * SCAN_CHUNK + (size_t)threadIdx.x * 4;
    unsigned c[4], s = 0;
#pragma unroll
    for (int j = 0; j < 4; ++j) { c[j] = __popc(present[base + j]); s += c[j]; }
    unsigned run = block_excl_scan_256(s, lds) + blockBase[blockIdx.x];
#pragma unroll
    for (int j = 0; j < 4; ++j) { wordPrefix[base + j] = run; run += c[j]; }
}

// ---------------------------------------------------------------------------
// K4: for each set bit: rank -> write coalesced index pair; duplicate keys get
// their destination value row pre-zeroed (atomic accumulation target).
// ---------------------------------------------------------------------------
__global__ __launch_bounds__(256) void k_write_indices(const unsigned* __restrict__ present,
                                                       const unsigned* __restrict__ dup,
                                                       const unsigned* __restrict__ wordPrefix,
                                                       int* __restrict__ outIdx0,
                                                       int* __restrict__ outIdx1,
                                                       float* __restrict__ outVals) {
    size_t w = (size_t)blockIdx.x * 256 + threadIdx.x;
    unsigned m = present[w];
    unsigned d = dup[w];
    unsigned r = wordPrefix[w];
    while (m) {
        unsigned b = (unsigned)__ffs(m) - 1u;
        m &= m - 1u;
        unsigned k = ((unsigned)w << 5) | b;
        outIdx0[r] = (int)(k >> 13);       // k / 8192
        outIdx1[r] = (int)(k & 8191u);     // k % 8192
        if ((d >> b) & 1u) {
            float4* row = (float4*)(outVals + (size_t)r * DENSE);
            float4 z = make_float4(0.f, 0.f, 0.f, 0.f);
#pragma unroll
            for (int j = 0; j < 4; ++j) row[j] = z;
        }
        ++r;
    }
}

// ---------------------------------------------------------------------------
// K5: padded tail (slots >= n_unique): indices [8192, 0], zero value rows.
// ---------------------------------------------------------------------------
__global__ __launch_bounds__(256) void k_fill_tail(const unsigned* __restrict__ nUnique,
                                                   int* __restrict__ outIdx0,
                                                   int* __restrict__ outIdx1,
                                                   float* __restrict__ outVals) {
    size_t u = (size_t)blockIdx.x * 256 + threadIdx.x;
    if (u >= (size_t)(*nUnique)) {
        outIdx0[u] = DIMV;   // fill // dim1 == 8192
        outIdx1[u] = 0;      // fill %  dim1 == 0
        float4* row = (float4*)(outVals + u * DENSE);
        float4 z = make_float4(0.f, 0.f, 0.f, 0.f);
#pragma unroll
        for (int j = 0; j < 4; ++j) row[j] = z;
    }
}

// ---------------------------------------------------------------------------
// K6: the bandwidth kernel. One thread owns one 64B value row.
//  - async-copy the row global -> LDS (4x b128, ASYNCcnt pipelined)
//  - rank = L2-resident wordPrefix lookup + popcount
//  - unique row (~97%): async-store LDS -> scattered destination (no VGPRs)
//  - duplicate row: read LDS, global_atomic_add_f32 into pre-zeroed row
// ---------------------------------------------------------------------------
__global__ __launch_bounds__(256) void k_scatter_values(const int* __restrict__ idx0,
                                                        const int* __restrict__ idx1,
                                                        const float* __restrict__ values,
                                                        const unsigned* __restrict__ present,
                                                        const unsigned* __restrict__ dup,
                                                        const unsigned* __restrict__ wordPrefix,
                                                        float* __restrict__ outVals) {
    __shared__ float smem[256 * DENSE];   // 16 KB, one 64B row per thread
    const int tid = threadIdx.x;
    size_t i = (size_t)blockIdx.x * 256 + tid;

    unsigned k = (unsigned)idx0[i] * (unsigned)DIMV + (unsigned)idx1[i];
    unsigned w = k >> 5, b = k & 31u;

    // LDS byte address: flat shared pointer's low 32 bits are the LDS offset.
    unsigned l = (unsigned)(uintptr_t)(&smem[tid * DENSE]);
    unsigned long long g = (unsigned long long)(uintptr_t)(values + i * DENSE);

    // Issue 4 async b128 copies global -> LDS for this row.
    asm volatile("global_load_async_to_lds_b128 %0, %1, off"
                 :: "v"(l),       "v"(g)        : "memory");
    asm volatile("global_load_async_to_lds_b128 %0, %1, off"
                 :: "v"(l + 16u), "v"(g + 16ull) : "memory");
    asm volatile("global_load_async_to_lds_b128 %0, %1, off"
                 :: "v"(l + 32u), "v"(g + 32ull) : "memory");
    asm volatile("global_load_async_to_lds_b128 %0, %1, off"
                 :: "v"(l + 48u), "v"(g + 48ull) : "memory");

    // Overlap rank computation (L2-resident 8MB tables) with async copies.
    unsigned r = wordPrefix[w] + (unsigned)__popc(present[w] & ((1u << b) - 1u));
    bool isDup = (dup[w] >> b) & 1u;
    float* dst = outVals + (size_t)r * DENSE;

    asm volatile("s_wait_asynccnt 0x0" ::: "memory");

    if (isDup) {
        const float4* s4 = (const float4*)(&smem[tid * DENSE]);
#pragma unroll
        for (int j = 0; j < 4; ++j) {
            float4 v = s4[j];
            atomicAdd(dst + 4 * j + 0, v.x);
            atomicAdd(dst + 4 * j + 1, v.y);
            atomicAdd(dst + 4 * j + 2, v.z);
            atomicAdd(dst + 4 * j + 3, v.w);
        }
    } else {
        unsigned long long go = (unsigned long long)(uintptr_t)dst;
        asm volatile("global_store_async_from_lds_b128 %0, %1, off"
                     :: "v"(go),         "v"(l)        : "memory");
        asm volatile("global_store_async_from_lds_b128 %0, %1, off"
                     :: "v"(go + 16ull), "v"(l + 16u)  : "memory");
        asm volatile("global_store_async_from_lds_b128 %0, %1, off"
                     :: "v"(go + 32ull), "v"(l + 32u)  : "memory");
        asm volatile("global_store_async_from_lds_b128 %0, %1, off"
                     :: "v"(go + 48ull), "v"(l + 48u)  : "memory");
    }
    asm volatile("s_wait_asynccnt 0x0" ::: "memory");
}

// ---------------------------------------------------------------------------
// Host-side launch sequence (all on `stream`, scratch only in d_ws).
// ---------------------------------------------------------------------------
extern "C" void kernel_launch(void* const* d_in, const int* in_sizes, int n_in,
                              void* d_out, int out_size, void* d_ws, size_t ws_size,
                              hipStream_t stream) {
    (void)in_sizes; (void)n_in; (void)out_size; (void)ws_size;

    const int*   indices = (const int*)d_in[0];       // [2, NNZ]
    const float* values  = (const float*)d_in[1];     // [NNZ, 16]
    const int* idx0 = indices;
    const int* idx1 = indices + NNZ;

    // Output layout: int32 indices [2,NNZ] | f32 values [NNZ,16] | int32 n_unique
    int*   outIdx0 = (int*)d_out;
    int*   outIdx1 = outIdx0 + NNZ;
    float* outVals = (float*)d_out + (size_t)2 * NNZ;
    int*   outN    = (int*)(outVals + (size_t)NNZ * DENSE);

    // Workspace layout (~24 MB; presence/dup/prefix tables fit in 192MB L2).
    unsigned* present    = (unsigned*)d_ws;            // WORDS
    unsigned* dup        = present + WORDS;            // WORDS
    unsigned* wordPrefix = dup + WORDS;                // WORDS
    unsigned* blockSums  = wordPrefix + WORDS;         // SCAN_BLKS
    unsigned* blockBase  = blockSums + SCAN_BLKS;      // SCAN_BLKS
    unsigned* nUnique    = blockBase + SCAN_BLKS;      // 1

    // K1: clear both bitmaps (2*WORDS u32, 4 per thread).
    k_clear<<<(2 * WORDS) / (256 * 4), 256, 0, stream>>>(present);

    // K2: mark presence + duplicates.
    k_mark<<<NNZ / 256, 256, 0, stream>>>(idx0, idx1, present, dup);

    // K3: hierarchical exclusive prefix-popcount over the bitmap.
    k_scan_partial<<<SCAN_BLKS, 256, 0, stream>>>(present, blockSums);
    k_scan_top<<<1, 256, 0, stream>>>(blockSums, blockBase, nUnique, outN);
    k_scan_down<<<SCAN_BLKS, 256, 0, stream>>>(present, blockBase, wordPrefix);

    // K4: unique -> coalesced indices; pre-zero duplicate-target value rows.
    k_write_indices<<<WORDS / 256, 256, 0, stream>>>(present, dup, wordPrefix,
                                                     outIdx0, outIdx1, outVals);

    // K5: padded tail slots.
    k_fill_tail<<<NNZ / 256, 256, 0, stream>>>(nUnique, outIdx0, outIdx1, outVals);

    // K6: async gather/scatter of the 64B value rows.
    k_scatter_values<<<NNZ / 256, 256, 0, stream>>>(idx0, idx1, values,
                                                    present, dup, wordPrefix, outVals);
}